// PostProcessor_36240934044432
// MI455X (gfx1250) — compile-verified
//
#include <hip/hip_runtime.h>
#include <math.h>

#define B_N   32
#define NOBJ  128
#define NREL  8192
#define OBJC  151
#define RELC  51
#define NMS_T 0.3f

typedef float v2f __attribute__((ext_vector_type(2)));
typedef float v8f __attribute__((ext_vector_type(8)));

// ---------------------------------------------------------------------------
// K1: per-(batch, class) 128x128 IoU >= 0.3 bitmask precompute.
// union-sum (area_i + area_j) for each 16x16 tile computed via
// v_wmma_f32_16x16x4_f32 (A = [areas | 1 | 0 | 0], B = [1 ; areas ; 0 ; 0]).
// mask layout: masks[(b*151+c)*128 + j][word w] bit x == overlap(i=32w+x, j)
// (IoU is symmetric so row/col orientation is interchangeable).
// ---------------------------------------------------------------------------
__global__ __launch_bounds__(32) void overlap_mask_kernel(
    const float* __restrict__ boxes, unsigned int* __restrict__ masks) {
  const int bc   = blockIdx.x;          // b*OBJC + c
  const int b    = bc / OBJC;
  const int c    = bc % OBJC;
  const int lane = threadIdx.x;         // 0..31, one full wave32

  __shared__ float4       box_s[NOBJ];
  __shared__ float        area_s[NOBJ];
  __shared__ unsigned int mask_s[NOBJ * 4];

  for (int j = lane; j < NOBJ; j += 32) {
    const float* p = boxes + (((size_t)b * NOBJ + j) * OBJC + c) * 4;
    float4 v; v.x = p[0]; v.y = p[1]; v.z = p[2]; v.w = p[3];
    box_s[j]  = v;
    area_s[j] = (v.z - v.x + 1.0f) * (v.w - v.y + 1.0f);
  }
  for (int w = lane; w < NOBJ * 4; w += 32) mask_s[w] = 0u;
  __syncthreads();

  const int hi = lane >> 4;   // 0: lanes 0-15 (K=0,1)   1: lanes 16-31 (K=2,3)
  const int nn = lane & 15;

  for (int t = 0; t < 64; ++t) {        // 8x8 grid of 16x16 tiles (uniform loop)
    const int i0 = (t >> 3) << 4;
    const int j0 = (t & 7) << 4;
    // A (16x4): col0 = area_i, col1 = 1 ;  B (4x16): row0 = 1, row1 = area_j
    v2f a, bv;
    a.x  = hi ? 0.0f : area_s[i0 + nn];
    a.y  = hi ? 0.0f : 1.0f;
    bv.x = hi ? 0.0f : 1.0f;
    bv.y = hi ? 0.0f : area_s[j0 + nn];
    v8f acc = {0.f, 0.f, 0.f, 0.f, 0.f, 0.f, 0.f, 0.f};
    acc = __builtin_amdgcn_wmma_f32_16x16x4_f32(
        false, a, false, bv, (short)0, acc, false, false);
    // lane holds D[m][n] for n = nn, m = v + 8*hi  ->  area_i + area_j
    const int    j  = j0 + nn;
    const float4 bj = box_s[j];
    unsigned int bits = 0u;
#pragma unroll
    for (int v = 0; v < 8; ++v) {
      const int    i  = i0 + v + (hi << 3);
      const float4 bi = box_s[i];
      float iw = fminf(bi.z, bj.z) - fmaxf(bi.x, bj.x) + 1.0f;
      float ih = fminf(bi.w, bj.w) - fmaxf(bi.y, bj.y) + 1.0f;
      iw = fmaxf(iw, 0.0f);
      ih = fmaxf(ih, 0.0f);
      const float inter = iw * ih;
      const float uni   = acc[v] - inter;      // area_i + area_j - inter
      if (inter / uni >= NMS_T) bits |= 1u << (i & 31);
    }
    // all 8 bits of this lane fall in the same 32-bit word (i0 mult of 16)
    atomicOr(&mask_s[j * 4 + (i0 >> 5)], bits);
  }
  __syncthreads();

  unsigned int* out = masks + (size_t)bc * (NOBJ * 4);
  for (int w = lane; w < NOBJ * 4; w += 32) out[w] = mask_s[w];
}

// ---------------------------------------------------------------------------
// K2: greedy NMS label assignment, one block per batch, thread i == object i.
// Probabilities are never materialized: thread keeps (max, 1/denom) and a
// 151-bit zero-mask; per-row best is cached and only rescanned when the
// suppressed class was that row's current argmax.
// ---------------------------------------------------------------------------
__global__ __launch_bounds__(NOBJ) void nms_kernel(
    const float* __restrict__ obj_logits, const float* __restrict__ boxes,
    const unsigned int* __restrict__ masks, float* __restrict__ out_boxes,
    float* __restrict__ out_scores, float* __restrict__ out_pred) {
  const int b = blockIdx.x;
  const int i = threadIdx.x;
  const float* lrow = obj_logits + ((size_t)b * NOBJ + i) * OBJC;

  float mx = -3.4e38f;
  for (int c = 0; c < OBJC; ++c) mx = fmaxf(mx, lrow[c]);
  float den = 0.0f;
  for (int c = 0; c < OBJC; ++c) den += expf(lrow[c] - mx);
  const float inv = 1.0f / den;

  __shared__ float        rbv[NOBJ];
  __shared__ int          rbf[NOBJ];
  __shared__ float        redv[NOBJ];
  __shared__ int          redf[NOBJ];
  __shared__ unsigned int zm[NOBJ][5];
  __shared__ int          label_s[NOBJ];
  __shared__ int          sel_s;

  for (int w = 0; w < 5; ++w) zm[i][w] = 0u;
  label_s[i] = 0;
  {
    float bv = -1.0f; int bc = 0;
    for (int c = 0; c < OBJC; ++c) {             // first-occurrence argmax
      const float p = (c == 0) ? 0.0f : expf(lrow[c] - mx) * inv;
      if (p > bv) { bv = p; bc = c; }
    }
    rbv[i] = bv;
    rbf[i] = i * OBJC + bc;                      // flat index tie-break
  }
  __syncthreads();

  for (int iter = 0; iter < NOBJ; ++iter) {
    redv[i] = rbv[i]; redf[i] = rbf[i];
    __syncthreads();
    for (int s = NOBJ / 2; s > 0; s >>= 1) {     // max, ties -> smaller flat
      if (i < s) {
        const float v2 = redv[i + s]; const int f2 = redf[i + s];
        if (v2 > redv[i] || (v2 == redv[i] && f2 < redf[i])) {
          redv[i] = v2; redf[i] = f2;
        }
      }
      __syncthreads();
    }
    if (i == 0) {
      sel_s = redf[0];
      label_s[redf[0] / OBJC] = redf[0] % OBJC;
    }
    __syncthreads();
    const int sb = sel_s / OBJC;
    const int sc = sel_s % OBJC;
    if (i == sb) {
      rbv[i] = -2.0f;                            // processed: never wins again
    } else if (rbv[i] >= 0.0f) {                 // only live rows
      const unsigned int mw =
          masks[(((size_t)b * OBJC + sc) * NOBJ + i) * 4 + (sb >> 5)];
      if ((mw >> (sb & 31)) & 1u) {
        zm[i][sc >> 5] |= 1u << (sc & 31);
        if (rbf[i] % OBJC == sc) {               // cached best got zeroed
          float bv = -1.0f; int bc = 0;
          for (int c = 0; c < OBJC; ++c) {
            const bool z = (zm[i][c >> 5] >> (c & 31)) & 1u;
            const float p = (c == 0 || z) ? 0.0f : expf(lrow[c] - mx) * inv;
            if (p > bv) { bv = p; bc = c; }
          }
          rbv[i] = bv; rbf[i] = i * OBJC + bc;
        }
      }
    }
    __syncthreads();
  }

  const int   c   = label_s[i];
  const float sco = (c == 0) ? 0.0f : expf(lrow[c] - mx) * inv;
  out_scores[b * NOBJ + i] = sco;
  out_pred[b * NOBJ + i]   = (float)c;
  const float* bp = boxes + (((size_t)b * NOBJ + i) * OBJC + c) * 4;
  float* ob = out_boxes + ((size_t)b * NOBJ + i) * 4;
  ob[0] = bp[0]; ob[1] = bp[1]; ob[2] = bp[2]; ob[3] = bp[3];
}

// ---------------------------------------------------------------------------
// K3: relation softmax + triple score + stable descending sort (bitonic in
// 64KB LDS) + gathered outputs. key = float_bits(triple)<<32 | (8191-r)<<6 |
// class  -> primary: triple desc (nonneg float bits monotone as uint),
// secondary: original index asc (matches JAX stable argsort).
// ---------------------------------------------------------------------------
__global__ __launch_bounds__(256) void rel_kernel(
    const float* __restrict__ rel_logits, const int* __restrict__ rel_pairs,
    const float* __restrict__ obj_scores, float* __restrict__ out_pair,
    float* __restrict__ out_prob, float* __restrict__ out_cls,
    float* __restrict__ out_tri) {
  const int b = blockIdx.x;
  const int t = threadIdx.x;
  __shared__ unsigned long long keys[NREL];      // 64 KB
  const float* rl = rel_logits + (size_t)b * NREL * RELC;
  const int*   rp = rel_pairs + (size_t)b * NREL * 2;
  const float* os = obj_scores + (size_t)b * NOBJ;

  for (int r = t; r < NREL; r += 256) {
    const float* row = rl + (size_t)r * RELC;
    float mx = row[0];
    for (int c = 1; c < RELC; ++c) mx = fmaxf(mx, row[c]);
    float den = 0.0f;
    for (int c = 0; c < RELC; ++c) den += expf(row[c] - mx);
    float bv = -1.0f; int bc = 1;
    for (int c = 1; c < RELC; ++c) {             // first-occurrence argmax
      const float p = expf(row[c] - mx) / den;
      if (p > bv) { bv = p; bc = c; }
    }
    const float s0  = os[rp[r * 2 + 0]];
    const float s1  = os[rp[r * 2 + 1]];
    const float tri = bv * s0 * s1;
    keys[r] = ((unsigned long long)__float_as_uint(tri) << 32) |
              ((unsigned long long)(unsigned)(NREL - 1 - r) << 6) |
              (unsigned long long)(unsigned)bc;
  }
  __syncthreads();

  for (int k = 2; k <= NREL; k <<= 1) {
    for (int j = k >> 1; j > 0; j >>= 1) {
      for (int idx = t; idx < NREL; idx += 256) {
        const int p = idx ^ j;
        if (p > idx) {
          const unsigned long long a = keys[idx], c2 = keys[p];
          const bool asc = (idx & k) != 0;       // overall: descending
          const bool sw  = asc ? (a > c2) : (a < c2);
          if (sw) { keys[idx] = c2; keys[p] = a; }
        }
      }
      __syncthreads();
    }
  }

  for (int r = t; r < NREL; r += 256) {
    const unsigned long long key = keys[r];
    const float tri = __uint_as_float((unsigned)(key >> 32));
    const int   idx = (NREL - 1) - (int)((key >> 6) & (NREL - 1));
    const int   cls = (int)(key & 63u);
    out_pair[((size_t)b * NREL + r) * 2 + 0] = (float)rp[idx * 2 + 0];
    out_pair[((size_t)b * NREL + r) * 2 + 1] = (float)rp[idx * 2 + 1];
    out_cls[(size_t)b * NREL + r] = (float)cls;
    out_tri[(size_t)b * NREL + r] = tri;
    const float* row = rl + (size_t)idx * RELC;
    float mx = row[0];
    for (int c = 1; c < RELC; ++c) mx = fmaxf(mx, row[c]);
    float den = 0.0f;
    for (int c = 0; c < RELC; ++c) den += expf(row[c] - mx);
    float* op = out_prob + ((size_t)b * NREL + r) * RELC;
    for (int c = 0; c < RELC; ++c) op[c] = expf(row[c] - mx) / den;
  }
}

// ---------------------------------------------------------------------------
extern "C" void kernel_launch(void* const* d_in, const int* in_sizes, int n_in,
                              void* d_out, int out_size, void* d_ws,
                              size_t ws_size, hipStream_t stream) {
  (void)in_sizes; (void)n_in; (void)out_size; (void)ws_size;
  const float* rel_logits = (const float*)d_in[0];  // (32, 8192, 51)
  const float* obj_logits = (const float*)d_in[1];  // (32, 128, 151)
  const int*   rel_pairs  = (const int*)d_in[2];    // (32, 8192, 2)
  const float* boxes      = (const float*)d_in[3];  // (32, 128, 151, 4)

  float* out = (float*)d_out;
  float* out_boxes  = out;              // 32*128*4      = 16384
  float* out_scores = out + 16384;      // 32*128        = 4096
  float* out_pred   = out + 20480;      // 32*128        = 4096
  float* out_pair   = out + 24576;      // 32*8192*2     = 524288
  float* out_prob   = out + 548864;     // 32*8192*51    = 13369344
  float* out_cls    = out + 13918208;   // 32*8192       = 262144
  float* out_tri    = out + 14180352;   // 32*8192       = 262144

  unsigned int* masks = (unsigned int*)d_ws;  // 32*151*128*4 u32 = ~9.9 MB

  overlap_mask_kernel<<<B_N * OBJC, 32, 0, stream>>>(boxes, masks);
  nms_kernel<<<B_N, NOBJ, 0, stream>>>(obj_logits, boxes, masks, out_boxes,
                                       out_scores, out_pred);
  rel_kernel<<<B_N, 256, 0, stream>>>(rel_logits, rel_pairs, out_scores,
                                      out_pair, out_prob, out_cls, out_tri);
}